// MS_DGF_GNN_38087769981230
// MI455X (gfx1250) — compile-verified
//
#include <hip/hip_runtime.h>
#include <hip/hip_bf16.h>

// ---------------------------------------------------------------------------
// MS_DGF_GNN for MI455X (gfx1250): f32 WMMA (16x16x4) for Gram + adj@x GEMMs,
// VALU for the per-pair sigma MLP (tanh-dominated).
// B=8, N=512, IN_DIM=4, H=128.
// Round 2: Gram A-fragments hoisted to registers (J-loop invariant),
//          v2f (b64) fragment loads from LDS.
// ---------------------------------------------------------------------------

#define B_  8
#define N_  512
#define H_  128

typedef float v2f __attribute__((ext_vector_type(2)));
typedef float v8f __attribute__((ext_vector_type(8)));

struct DgfHeads {
    const float* W1[4];   // [256,32] row-major
    const float* b1[4];   // [32]
    const float* W2[4];   // [32]
    const float* b2[4];   // [1]
};

__device__ __forceinline__ v8f wmma_f32_4(v2f a, v2f b, v8f c) {
    // D = A(16x4,f32) * B(4x16,f32) + C(16x16,f32)
    return __builtin_amdgcn_wmma_f32_16x16x4_f32(
        /*neg_a=*/false, a, /*neg_b=*/false, b,
        /*c_mod=*/(short)0, c, /*reuse_a=*/false, /*reuse_b=*/false);
}

__device__ __forceinline__ float elu_f(float v) {
    return v > 0.f ? v : (__expf(v) - 1.f);
}

// ---------------------------------------------------------------------------
// 1) feat_ext: h = ELU(LN(x @ feW + feb)); one block (128 thr) per (b,n) row
// ---------------------------------------------------------------------------
__global__ void k_featext(const float* __restrict__ x,
                          const float* __restrict__ feW, const float* __restrict__ feb,
                          const float* __restrict__ lng, const float* __restrict__ lnb,
                          float* __restrict__ hout) {
    const int bn = blockIdx.x;
    const int j  = threadIdx.x;                 // 0..127
    __shared__ float xr[4];
    __shared__ float red[H_];
    if (j < 4) xr[j] = x[(size_t)bn * 4 + j];
    __syncthreads();
    float v = feb[j];
#pragma unroll
    for (int k = 0; k < 4; ++k) v += xr[k] * feW[k * H_ + j];
    red[j] = v;
    __syncthreads();
    for (int s = 64; s > 0; s >>= 1) { if (j < s) red[j] += red[j + s]; __syncthreads(); }
    const float mu = red[0] * (1.f / H_);
    __syncthreads();
    const float d0 = v - mu;
    red[j] = d0 * d0;
    __syncthreads();
    for (int s = 64; s > 0; s >>= 1) { if (j < s) red[j] += red[j + s]; __syncthreads(); }
    const float var = red[0] * (1.f / H_);
    const float hn  = d0 * rsqrtf(var + 1e-5f) * lng[j] + lnb[j];
    hout[(size_t)bn * H_ + j] = elu_f(hn);
}

// ---------------------------------------------------------------------------
// 2) channel attention weights: w = sigmoid(relu(mean_n(h) @ W1 + b1) @ W2 + b2)
// ---------------------------------------------------------------------------
__global__ void k_chatt(const float* __restrict__ h,
                        const float* __restrict__ caW1, const float* __restrict__ cab1,
                        const float* __restrict__ caW2, const float* __restrict__ cab2,
                        float* __restrict__ wout) {
    const int b = blockIdx.x;
    const int j = threadIdx.x;                  // 0..127
    __shared__ float m[H_];
    __shared__ float a1[16];
    float s = 0.f;
    for (int n = 0; n < N_; ++n) s += h[((size_t)b * N_ + n) * H_ + j];
    m[j] = s * (1.f / N_);
    __syncthreads();
    if (j < 16) {
        float a = cab1[j];
        for (int k = 0; k < H_; ++k) a += m[k] * caW1[k * 16 + j];
        a1[j] = fmaxf(a, 0.f);
    }
    __syncthreads();
    float z = cab2[j];
#pragma unroll
    for (int k = 0; k < 16; ++k) z += a1[k] * caW2[k * H_ + j];
    wout[(size_t)b * H_ + j] = 1.f / (1.f + __expf(-z));
}

__global__ void k_scale(float* __restrict__ h, const float* __restrict__ w) {
    const size_t i = (size_t)blockIdx.x * blockDim.x + threadIdx.x;   // B*N*H
    const int j = (int)(i & (H_ - 1));
    const size_t b = i >> 16;                   // / (512*128)
    h[i] *= w[b * H_ + j];
}

// ---------------------------------------------------------------------------
// 3) per-row features: sq[b,n] = ||h||^2 ; hi = h@W1[:128], hj = h@W1[128:]
//    one block (256 thr) per (b,n)
// ---------------------------------------------------------------------------
__global__ void k_rowfeat(const float* __restrict__ h_in, DgfHeads heads, int nh,
                          float* __restrict__ hiF, float* __restrict__ hjF,
                          float* __restrict__ sq) {
    const int bn = blockIdx.x;
    const int t  = threadIdx.x;                 // 0..255
    __shared__ float row[H_];
    __shared__ float red[H_];
    if (t < H_) row[t] = h_in[(size_t)bn * H_ + t];
    __syncthreads();
    if (t < H_) red[t] = row[t] * row[t];
    __syncthreads();
    for (int s = 64; s > 0; s >>= 1) { if (t < s) red[t] += red[t + s]; __syncthreads(); }
    if (t == 0) sq[bn] = red[0];
    if (t < nh * 64) {
        const int hd    = t >> 6;
        const int rem   = t & 63;
        const int which = rem >> 5;             // 0: hi, 1: hj
        const int k     = rem & 31;
        const float* W1 = heads.W1[hd] + (which ? H_ * 32 : 0);
        float s = 0.f;
        for (int c = 0; c < H_; ++c) s += row[c] * W1[c * 32 + k];
        float* dst = which ? hjF : hiF;
        const int b = bn >> 9, n = bn & (N_ - 1);
        dst[(((size_t)b * 4 + hd) * N_ + n) * 32 + k] = s;
    }
}

// ---------------------------------------------------------------------------
// 4) DGF tile kernel: one wave per (b, I-tile). For each J-tile:
//      gram = H0I @ H0J^T           (32x v_wmma_f32_16x16x4_f32, K=128)
//      adj  = mean_h exp(-dist/(2*sigma_h^2+eps))   (VALU tanh MLP)
//      acc += adj @ H0J             (32x v_wmma_f32_16x16x4_f32)
//    Gram A-fragments are J-invariant: kept in 64 VGPRs, loaded once from
//    global as aligned b64s; B/adj fragments are aligned v2f LDS reads.
// ---------------------------------------------------------------------------
__global__ void __launch_bounds__(32)
k_dgf(const float* __restrict__ h_in, const float* __restrict__ hiF,
      const float* __restrict__ hjF,  const float* __restrict__ sq,
      DgfHeads heads, int nh, float* __restrict__ tbuf) {
    const int b    = blockIdx.x;
    const int I    = blockIdx.y * 16;
    const int lane = threadIdx.x;               // 0..31
    const int nl   = lane & 15;                 // M (A) / N (B,C) index
    const int hl   = lane >> 4;                 // lane half

    __shared__ float sH0J[16][H_];
    __shared__ float sHiI[4][16][32];
    __shared__ float sHjJ[4][16][32];
    __shared__ float sAdj[16][16];
    __shared__ float sSqI[16], sSqJ[16];
    __shared__ float sB1[4][32], sW2[4][32], sB2[4];

    // ---- J-invariant state -------------------------------------------------
    // Gram A fragments: lane owns row I+nl; frag t covers K = 4t..4t+3,
    // this lane holds elements {4t+2*hl, 4t+2*hl+1} (8-byte aligned pairs).
    v2f aI[32];
    {
        const float* arow = h_in + ((size_t)b * N_ + I + nl) * H_ + 2 * hl;
#pragma unroll
        for (int t = 0; t < 32; ++t)
            aI[t] = *(const v2f*)(arow + 4 * t);
    }
    for (int hd = 0; hd < nh; ++hd) {
        const float* hi = hiF + (((size_t)b * 4 + hd) * N_ + I) * 32;
        for (int t = lane; t < 16 * 32; t += 32) sHiI[hd][t >> 5][t & 31] = hi[t];
        sB1[hd][lane] = heads.b1[hd][lane];
        sW2[hd][lane] = heads.W2[hd][lane];
        if (lane == 0) sB2[hd] = heads.b2[hd][0];
    }
    if (lane < 16) sSqI[lane] = sq[(size_t)b * N_ + I + lane];

    v8f acc[8];
    const v8f vzero = {0.f, 0.f, 0.f, 0.f, 0.f, 0.f, 0.f, 0.f};
#pragma unroll
    for (int c = 0; c < 8; ++c) acc[c] = vzero;

    const float inv_nh = 1.f / (float)nh;

    for (int J = 0; J < N_; J += 16) {
        __syncthreads();                        // protect sH0J/sHjJ reuse
        const float* hbj = h_in + ((size_t)b * N_ + J) * H_;
        for (int t = lane; t < 16 * H_; t += 32) sH0J[t >> 7][t & (H_ - 1)] = hbj[t];
        for (int hd = 0; hd < nh; ++hd) {
            const float* hj = hjF + (((size_t)b * 4 + hd) * N_ + J) * 32;
            for (int t = lane; t < 16 * 32; t += 32) sHjJ[hd][t >> 5][t & 31] = hj[t];
        }
        if (lane < 16) sSqJ[lane] = sq[(size_t)b * N_ + J + lane];
        __syncthreads();

        // ---- Gram tile: A (regs) x B = H0J^T (4x16 slices from LDS) ----
        v8f g = vzero;
#pragma unroll
        for (int t = 0; t < 32; ++t) {
            const v2f bb = *(const v2f*)&sH0J[nl][4 * t + 2 * hl];
            g = wmma_f32_4(aI[t], bb, g);
        }

        // ---- per-pair sigma MLP + Gaussian adjacency (VALU) ----
        float adjr[8];
#pragma unroll
        for (int r = 0; r < 8; ++r) {
            const int m = r + 8 * hl;           // C-layout row for VGPR r
            float dist = sSqI[m] + sSqJ[nl] - 2.f * g[r];
            dist = fmaxf(dist, 0.f);
            float asum = 0.f;
            for (int hd = 0; hd < nh; ++hd) {
                float s = sB2[hd];
#pragma unroll
                for (int k = 0; k < 32; ++k) {
                    const float pre = sHiI[hd][m][k] + sHjJ[hd][nl][k] + sB1[hd][k];
                    s += tanhf(pre) * sW2[hd][k];
                }
                const float sig = (s > 20.f) ? s : log1pf(__expf(s));   // softplus
                asum += __expf(-dist / (2.f * sig * sig + 1e-6f));
            }
            adjr[r] = asum * inv_nh;
        }
        // bounce adj through LDS to re-fragment as an A matrix
#pragma unroll
        for (int r = 0; r < 8; ++r) sAdj[r + 8 * hl][nl] = adjr[r];
        __syncthreads();

        // ---- acc += adj(16x16) @ H0J(16x128) ----
#pragma unroll
        for (int c = 0; c < 8; ++c) {
#pragma unroll
            for (int kt = 0; kt < 4; ++kt) {
                const int k = 4 * kt + 2 * hl;
                const v2f a = *(const v2f*)&sAdj[nl][k];
                v2f bb;
                bb.x = sH0J[k][16 * c + nl];
                bb.y = sH0J[k + 1][16 * c + nl];
                acc[c] = wmma_f32_4(a, bb, acc[c]);
            }
        }
    }

    // store t rows (C layout: VGPR r -> row r + 8*hl, col = 16*c + nl)
    float* tb = tbuf + ((size_t)b * N_ + I) * H_;
#pragma unroll
    for (int c = 0; c < 8; ++c)
#pragma unroll
        for (int r = 0; r < 8; ++r)
            tb[(r + 8 * hl) * H_ + 16 * c + nl] = acc[c][r];
}

// ---------------------------------------------------------------------------
// 5) projection + residual + ELU: hout = ELU(t @ pW + pb + hres)
// ---------------------------------------------------------------------------
__global__ void k_proj_res(const float* __restrict__ t, const float* __restrict__ pW,
                           const float* __restrict__ pb, const float* __restrict__ hres,
                           float* __restrict__ hout) {
    const int bn = blockIdx.x;
    const int j  = threadIdx.x;                 // 0..127
    __shared__ float tr[H_];
    tr[j] = t[(size_t)bn * H_ + j];
    __syncthreads();
    float s = pb[j];
    for (int k = 0; k < H_; ++k) s += tr[k] * pW[k * H_ + j];
    s += hres[(size_t)bn * H_ + j];
    hout[(size_t)bn * H_ + j] = elu_f(s);
}

// ---------------------------------------------------------------------------
// 6) pooling + classifier head; one block (128 thr) per batch
// ---------------------------------------------------------------------------
__global__ void k_head(const float* __restrict__ h2,
                       const float* __restrict__ clW1, const float* __restrict__ clb1,
                       const float* __restrict__ bng,  const float* __restrict__ bnb,
                       const float* __restrict__ bnm,  const float* __restrict__ bnv,
                       const float* __restrict__ clW2, const float* __restrict__ clb2,
                       float* __restrict__ out) {
    const int b = blockIdx.x;
    const int j = threadIdx.x;                  // 0..127
    __shared__ float feat[2 * H_];
    __shared__ float z[64];
    float mn = 0.f, mx = -3.402823e38f;
    for (int n = 0; n < N_; ++n) {
        const float v = h2[((size_t)b * N_ + n) * H_ + j];
        mn += v; mx = fmaxf(mx, v);
    }
    feat[j] = mn * (1.f / N_);
    feat[H_ + j] = mx;
    __syncthreads();
    if (j < 64) {
        float s = clb1[j];
        for (int k = 0; k < 2 * H_; ++k) s += feat[k] * clW1[k * 64 + j];
        s = (s - bnm[j]) * rsqrtf(bnv[j] + 1e-5f) * bng[j] + bnb[j];
        z[j] = s >= 0.f ? s : 0.2f * s;
    }
    __syncthreads();
    if (j < 2) {
        float s = clb2[j];
#pragma unroll
        for (int k = 0; k < 64; ++k) s += z[k] * clW2[k * 2 + j];
        out[b * 2 + j] = s;
    }
}

// ---------------------------------------------------------------------------
// launch
// ---------------------------------------------------------------------------
extern "C" void kernel_launch(void* const* d_in, const int* in_sizes, int n_in,
                              void* d_out, int out_size, void* d_ws, size_t ws_size,
                              hipStream_t stream) {
    (void)in_sizes; (void)n_in; (void)out_size; (void)ws_size;
    // input order = setup_inputs() insertion order, flattened
    const float* x    = (const float*)d_in[0];
    const float* feW  = (const float*)d_in[1];
    const float* feb  = (const float*)d_in[2];
    const float* lng  = (const float*)d_in[3];
    const float* lnb  = (const float*)d_in[4];
    const float* caW1 = (const float*)d_in[5];
    const float* cab1 = (const float*)d_in[6];
    const float* caW2 = (const float*)d_in[7];
    const float* cab2 = (const float*)d_in[8];

    DgfHeads hd1, hd2;
    for (int h = 0; h < 4; ++h) {
        hd1.W1[h] = (const float*)d_in[9 + 4 * h + 0];
        hd1.b1[h] = (const float*)d_in[9 + 4 * h + 1];
        hd1.W2[h] = (const float*)d_in[9 + 4 * h + 2];
        hd1.b2[h] = (const float*)d_in[9 + 4 * h + 3];
    }
    const float* g1pW = (const float*)d_in[25];
    const float* g1pb = (const float*)d_in[26];
    for (int h = 0; h < 2; ++h) {
        hd2.W1[h] = (const float*)d_in[27 + 4 * h + 0];
        hd2.b1[h] = (const float*)d_in[27 + 4 * h + 1];
        hd2.W2[h] = (const float*)d_in[27 + 4 * h + 2];
        hd2.b2[h] = (const float*)d_in[27 + 4 * h + 3];
    }
    hd2.W1[2] = hd2.W1[0]; hd2.b1[2] = hd2.b1[0]; hd2.W2[2] = hd2.W2[0]; hd2.b2[2] = hd2.b2[0];
    hd2.W1[3] = hd2.W1[1]; hd2.b1[3] = hd2.b1[1]; hd2.W2[3] = hd2.W2[1]; hd2.b2[3] = hd2.b2[1];
    const float* g2pW = (const float*)d_in[35];
    const float* g2pb = (const float*)d_in[36];
    const float* clW1 = (const float*)d_in[37];
    const float* clb1 = (const float*)d_in[38];
    const float* bng  = (const float*)d_in[39];
    const float* bnb  = (const float*)d_in[40];
    const float* bnm  = (const float*)d_in[41];
    const float* bnv  = (const float*)d_in[42];
    const float* clW2 = (const float*)d_in[43];
    const float* clb2 = (const float*)d_in[44];

    float* ws = (float*)d_ws;
    const size_t SZ = (size_t)B_ * N_ * H_;     // 524288
    float* H0 = ws;                             // h (in-place scaled), later h2
    float* H1 = ws + SZ;
    float* T  = ws + 2 * SZ;
    float* HI = ws + 3 * SZ;                    // B*4*N*32 = SZ
    float* HJ = ws + 4 * SZ;
    float* SQ = ws + 5 * SZ;                    // B*N
    float* W  = SQ + (size_t)B_ * N_;           // B*H

    k_featext<<<B_ * N_, H_, 0, stream>>>(x, feW, feb, lng, lnb, H0);
    k_chatt<<<B_, H_, 0, stream>>>(H0, caW1, cab1, caW2, cab2, W);
    k_scale<<<(int)(SZ / 256), 256, 0, stream>>>(H0, W);

    // DGF block 1 (4 heads)
    k_rowfeat<<<B_ * N_, 256, 0, stream>>>(H0, hd1, 4, HI, HJ, SQ);
    k_dgf<<<dim3(B_, N_ / 16), 32, 0, stream>>>(H0, HI, HJ, SQ, hd1, 4, T);
    k_proj_res<<<B_ * N_, H_, 0, stream>>>(T, g1pW, g1pb, H0, H1);

    // DGF block 2 (2 heads)
    k_rowfeat<<<B_ * N_, 256, 0, stream>>>(H1, hd2, 2, HI, HJ, SQ);
    k_dgf<<<dim3(B_, N_ / 16), 32, 0, stream>>>(H1, HI, HJ, SQ, hd2, 2, T);
    k_proj_res<<<B_ * N_, H_, 0, stream>>>(T, g2pW, g2pb, H1, H0);   // h2 -> H0

    k_head<<<B_, H_, 0, stream>>>(H0, clW1, clb1, bng, bnb, bnm, bnv, clW2, clb2,
                                  (float*)d_out);
}